// MultiHeadAttention_graph_30064771072763
// MI455X (gfx1250) — compile-verified
//
#include <hip/hip_runtime.h>
#include <hip/hip_bf16.h>

// MI455X (gfx1250) implementation. f32 end-to-end (memory-bound: 512MB of
// vec_adj streaming dominates; compute precision is free), all matmuls via
// V_WMMA_F32_16X16X4_F32. wave32; 256-thread blocks = 8 waves.
// vec_adj tile staging uses the Tensor Data Mover (TENSOR_LOAD_TO_LDS with
// pad_enable for a conflict-free LDS stride), tracked on TENSORcnt.

typedef __attribute__((ext_vector_type(2))) float v2f;
typedef __attribute__((ext_vector_type(8))) float v8f;
typedef __attribute__((ext_vector_type(4))) unsigned int v4u;
typedef __attribute__((ext_vector_type(4))) int v4i;
typedef __attribute__((ext_vector_type(8))) int v8i;

#define LSA_STRIDE 20   // 64x16 A tile, padded to dodge bank conflicts
#define LSB_STRIDE 80   // 16x64 B tile, padded so lane halves hit disjoint banks

#if defined(__has_builtin)
#if __has_builtin(__builtin_amdgcn_tensor_load_to_lds) && \
    __has_builtin(__builtin_amdgcn_s_wait_tensorcnt)
#define USE_TDM 1
#endif
#endif

__device__ __forceinline__ v8f wmma4(v2f a, v2f b, v8f c) {
  // D = A(16x4,f32) * B(4x16,f32) + C(16x16,f32)
  return __builtin_amdgcn_wmma_f32_16x16x4_f32(false, a, false, b, (short)0, c,
                                               false, false);
}

__device__ __forceinline__ float gelu_tanh_f(float x) {
  float u = 0.7978845608028654f * (x + 0.044715f * x * x * x);
  return 0.5f * x * (1.0f + tanhf(u));
}

#ifdef USE_TDM
// 2-D TDM tile load: f32 elements, row-major tile [tile_d1 rows x tile_d0
// cols], row stride d0_stride elems; LDS padding: +4 dwords per 64 dwords
// stored => LDS row stride = tile_d0 + 4 = 68 floats for tile_d0 == 64.
__device__ __forceinline__ void tdm_load_2d(unsigned lds_off, const void* gptr,
                                            unsigned tensor_d0, unsigned tensor_d1,
                                            unsigned tile_d0, unsigned tile_d1,
                                            unsigned d0_stride) {
  unsigned long long ga = (unsigned long long)(size_t)gptr;
  v4u g0;
  g0[0] = 1u;                                   // count=1, user mode, no gather
  g0[1] = lds_off;                              // LDS byte address
  g0[2] = (unsigned)(ga & 0xffffffffull);       // global addr [31:0]
  g0[3] = (unsigned)((ga >> 32) & 0x1ffffffull) // global addr [56:32]
          | (2u << 30);                         // type = 2 ("image")
  v8i g1;
  // wg_mask=0 | data_size=2 (4B) | pad_enable | pad_interval=5 (64 dwords)
  // | pad_amount=3 (4 dwords)
  g1[0] = (2 << 16) | (1 << 20) | (5 << 22) | (3 << 25);
  g1[1] = (int)((tensor_d0 & 0xffffu) << 16);   // tensor_dim0[15:0] @ bits 63:48
  g1[2] = (int)(((tensor_d0 >> 16) & 0xffffu) | ((tensor_d1 & 0xffffu) << 16));
  g1[3] = (int)(((tensor_d1 >> 16) & 0xffffu) | ((tile_d0 & 0xffffu) << 16));
  g1[4] = (int)(tile_d1 & 0xffffu);             // tile_dim1; tile_dim2 = 0
  g1[5] = (int)d0_stride;                       // tensor_dim0_stride[31:0]
  g1[6] = 0;                                    // stride hi / dim1_stride lo
  g1[7] = 0;
  v4i z4 = {0, 0, 0, 0};
  v8i z8 = {0, 0, 0, 0, 0, 0, 0, 0};
  // 6-arg form (clang-23 / therock-10.0 headers)
  __builtin_amdgcn_tensor_load_to_lds(g0, g1, z4, z4, z8, 0);
}
#endif

// ---------------------------------------------------------------------------
// Generic 64x64-block WMMA GEMM core. A row-major [M,K] (lda), B row-major
// [K,N] (ldb); caller passes block-offset base pointers. 8 waves: wave (wm,wn)
// owns a 16x32 strip (2 accumulators).
// ---------------------------------------------------------------------------
__device__ __forceinline__ void gemm_tile_f32(const float* __restrict__ A, int lda,
                                              const float* __restrict__ B, int ldb,
                                              int K, v8f acc[2],
                                              float* lsA, float* lsB) {
  const int t = threadIdx.x;
  const int lane = t & 31;
  const int wave = t >> 5;
  const int wm = wave >> 1;        // 0..3 (M strip)
  const int wn = wave & 1;         // 0..1 (N strip)
  const int r  = lane & 15;
  const int kb = (lane >> 4) * 2;  // K sub-offset per ISA A/B layout

  for (int k0 = 0; k0 < K; k0 += 16) {
    __syncthreads();
    { // A tile 64x16 (float4-coalesced)
      int row = t >> 2, c4 = (t & 3) * 4;
      float4 v = *(const float4*)(A + (long)row * lda + k0 + c4);
      float* d = lsA + row * LSA_STRIDE + c4;
      d[0] = v.x; d[1] = v.y; d[2] = v.z; d[3] = v.w;
    }
    { // B tile 16x64 (float4-coalesced)
      int kr = t >> 4, c4 = (t & 15) * 4;
      float4 v = *(const float4*)(B + (long)(k0 + kr) * ldb + c4);
      float* d = lsB + kr * LSB_STRIDE + c4;
      d[0] = v.x; d[1] = v.y; d[2] = v.z; d[3] = v.w;
    }
    if (k0 + 16 < K) { // gfx1250 global_prefetch_b8 of next K tile
      __builtin_prefetch(A + (long)(t >> 2) * lda + k0 + 16, 0, 0);
      __builtin_prefetch(B + (long)(k0 + 16 + (t >> 4)) * ldb + (t & 15) * 4, 0, 0);
    }
    __syncthreads();
#pragma unroll
    for (int kk = 0; kk < 16; kk += 4) {
      v2f a;
      a.x = lsA[(wm * 16 + r) * LSA_STRIDE + kk + kb];
      a.y = lsA[(wm * 16 + r) * LSA_STRIDE + kk + kb + 1];
#pragma unroll
      for (int nt = 0; nt < 2; ++nt) {
        int col = wn * 32 + nt * 16 + r;
        v2f b;
        b.x = lsB[(kk + kb) * LSB_STRIDE + col];
        b.y = lsB[(kk + kb + 1) * LSB_STRIDE + col];
        acc[nt] = wmma4(a, b, acc[nt]);
      }
    }
  }
}

// Batched GEMM: z -> (b = z>>3, h = z&7) with independent b/h strides.
__global__ __launch_bounds__(256) void k_gemm_batched(
    const float* __restrict__ A, long sAb, long sAh, int lda,
    const float* __restrict__ B, long sBb, long sBh, int ldb,
    float* __restrict__ C, long sCb, long sCh, int ldc,
    int K, float alpha) {
  __shared__ float lsA[64 * LSA_STRIDE];
  __shared__ float lsB[16 * LSB_STRIDE];
  const int z = blockIdx.z, bb = z >> 3, hh = z & 7;
  const float* Ab = A + bb * sAb + hh * sAh + (long)blockIdx.y * 64 * lda;
  const float* Bb = B + bb * sBb + hh * sBh + (long)blockIdx.x * 64;
  v8f acc[2] = {{0,0,0,0,0,0,0,0},{0,0,0,0,0,0,0,0}};
  gemm_tile_f32(Ab, lda, Bb, ldb, K, acc, lsA, lsB);
  const int t = threadIdx.x, lane = t & 31, wave = t >> 5;
  const int wm = wave >> 1, wn = wave & 1;
  const int hi = lane >> 4, col = lane & 15;
  float* Cb = C + bb * sCb + hh * sCh;
#pragma unroll
  for (int nt = 0; nt < 2; ++nt) {
    int cn = blockIdx.x * 64 + wn * 32 + nt * 16 + col;
#pragma unroll
    for (int rr = 0; rr < 8; ++rr) {
      int cm = blockIdx.y * 64 + wm * 16 + rr + 8 * hi;
      Cb[(long)cm * ldc + cn] = alpha * acc[nt][rr];
    }
  }
}

// FC GEMM with fused tanh-GELU + residual epilogue: qo = resid + gelu(A@B)
__global__ __launch_bounds__(256) void k_fc(
    const float* __restrict__ A, const float* __restrict__ B,
    const float* __restrict__ resid, float* __restrict__ C) {
  __shared__ float lsA[64 * LSA_STRIDE];
  __shared__ float lsB[16 * LSB_STRIDE];
  const float* Ab = A + (long)blockIdx.y * 64 * 512;
  const float* Bb = B + (long)blockIdx.x * 64;
  v8f acc[2] = {{0,0,0,0,0,0,0,0},{0,0,0,0,0,0,0,0}};
  gemm_tile_f32(Ab, 512, Bb, 512, 512, acc, lsA, lsB);
  const int t = threadIdx.x, lane = t & 31, wave = t >> 5;
  const int wm = wave >> 1, wn = wave & 1;
  const int hi = lane >> 4, col = lane & 15;
#pragma unroll
  for (int nt = 0; nt < 2; ++nt) {
    int cn = blockIdx.x * 64 + wn * 32 + nt * 16 + col;
#pragma unroll
    for (int rr = 0; rr < 8; ++rr) {
      long cm = blockIdx.y * 64 + wm * 16 + rr + 8 * hi;
      C[cm * 512 + cn] = resid[cm * 512 + cn] + gelu_tanh_f(acc[nt][rr]);
    }
  }
}

// GNN aggregation: x = self_info + ((nw[m]*adj[l,m]) @ dd_info) / denom[l]
__global__ __launch_bounds__(256) void k_agg(
    const int* __restrict__ adj, const float* __restrict__ nw,
    const float* __restrict__ dd_info, const float* __restrict__ self_info,
    const float* __restrict__ denom, float* __restrict__ xout) {
  __shared__ float lsA[64 * LSA_STRIDE];
  __shared__ float lsB[16 * LSB_STRIDE];
  const int b = blockIdx.z;
  const int t = threadIdx.x, lane = t & 31, wave = t >> 5;
  const int wm = wave >> 1, wn = wave & 1;
  const int r = lane & 15, kb = (lane >> 4) * 2;
  const int* adjb = adj + ((long)b * 512 + blockIdx.y * 64) * 512;
  const float* nwb = nw + b * 512;
  const float* Bb = dd_info + (long)b * 512 * 512 + blockIdx.x * 64;
  v8f acc[2] = {{0,0,0,0,0,0,0,0},{0,0,0,0,0,0,0,0}};
  for (int k0 = 0; k0 < 512; k0 += 16) {
    __syncthreads();
    { // A = nw[m] * adj[l,m]
      int row = t >> 2, c4 = (t & 3) * 4;
      int4 av = *(const int4*)(adjb + (long)row * 512 + k0 + c4);
      float* d = lsA + row * LSA_STRIDE + c4;
      d[0] = av.x ? nwb[k0 + c4 + 0] : 0.0f;
      d[1] = av.y ? nwb[k0 + c4 + 1] : 0.0f;
      d[2] = av.z ? nwb[k0 + c4 + 2] : 0.0f;
      d[3] = av.w ? nwb[k0 + c4 + 3] : 0.0f;
    }
    {
      int kr = t >> 4, c4 = (t & 15) * 4;
      float4 v = *(const float4*)(Bb + (long)(k0 + kr) * 512 + c4);
      float* d = lsB + kr * LSB_STRIDE + c4;
      d[0] = v.x; d[1] = v.y; d[2] = v.z; d[3] = v.w;
    }
    __syncthreads();
#pragma unroll
    for (int kk = 0; kk < 16; kk += 4) {
      v2f a;
      a.x = lsA[(wm * 16 + r) * LSA_STRIDE + kk + kb];
      a.y = lsA[(wm * 16 + r) * LSA_STRIDE + kk + kb + 1];
#pragma unroll
      for (int nt = 0; nt < 2; ++nt) {
        int col = wn * 32 + nt * 16 + r;
        v2f bv;
        bv.x = lsB[(kk + kb) * LSB_STRIDE + col];
        bv.y = lsB[(kk + kb + 1) * LSB_STRIDE + col];
        acc[nt] = wmma4(a, bv, acc[nt]);
      }
    }
  }
  const int hi = lane >> 4, col = lane & 15;
#pragma unroll
  for (int nt = 0; nt < 2; ++nt) {
    int cn = blockIdx.x * 64 + wn * 32 + nt * 16 + col;
#pragma unroll
    for (int rr = 0; rr < 8; ++rr) {
      long cm = (long)b * 512 + blockIdx.y * 64 + wm * 16 + rr + 8 * hi;
      xout[cm * 512 + cn] = self_info[cm * 512 + cn] + acc[nt][rr] / denom[cm];
    }
  }
}

// khT[b][h][d][m] = kh_p[b*512+m][h*64+d]  (so content scores use row-major B)
__global__ __launch_bounds__(256) void k_transpose_kh(
    const float* __restrict__ kh, float* __restrict__ khT) {
  long o = (long)blockIdx.x * 256 + threadIdx.x;  // 4*8*64*512 = 1048576
  int m = o & 511, d = (o >> 9) & 63, h = (o >> 15) & 7, b = (int)(o >> 18);
  khT[o] = kh[((long)(b * 512 + m)) * 512 + h * 64 + d];
}

// Relative scores: per (b,l): S_rel[h,m] = qs[h,:] . vec_adj_k[b,l,m,:]
// A = qs padded to 16 rows. B tile = vec_adj_k[b,l, m0:m0+128, 0:64] DMA'd
// into LDS by the TDM, row-major with pad => LDS stride 68 floats. The B
// fragment is read column-wise: B[k=d][n=m] = lsB[m*68 + d] (k-pair adjacent
// => single ds_load_b64; lane halves hit disjoint banks).
__global__ __launch_bounds__(256) void k_rel_scores(
    const float* __restrict__ qs_p, const float* __restrict__ vak,
    float* __restrict__ scores) {
  __shared__ float lsA[16 * 68];
  __shared__ float lsB[128 * 68];
  const int bl = blockIdx.x, b = bl >> 9, l = bl & 511;
  const int m0 = blockIdx.y * 128;
  const int t = threadIdx.x;
  const int lane = t & 31, wave = t >> 5;

#ifdef USE_TDM
  if (wave == 0) {
    tdm_load_2d((unsigned)(size_t)(void*)lsB,
                vak + ((long)bl * 512 + m0) * 64,
                /*tensor_d0=*/64, /*tensor_d1=*/512,
                /*tile_d0=*/64, /*tile_d1=*/128, /*d0_stride=*/64);
  }
#else
  { // manual fallback: 128 rows x 64 floats, float4-coalesced, stride-68 rows
    int ml = t >> 1, half = t & 1;
    const float* src = vak + ((long)bl * 512 + m0 + ml) * 64 + half * 32;
#pragma unroll
    for (int j = 0; j < 8; ++j) {
      float4 v = *(const float4*)(src + j * 4);
      *(float4*)(lsB + ml * 68 + half * 32 + j * 4) = v;
    }
  }
#endif
  { // A: rows h<8 real, 8..15 zero
    int row = t >> 4, c4 = (t & 15) * 4;
    float4 v = {0.f, 0.f, 0.f, 0.f};
    if (row < 8) v = *(const float4*)(qs_p + (long)bl * 512 + row * 64 + c4);
    lsA[row * 68 + c4 + 0] = v.x; lsA[row * 68 + c4 + 1] = v.y;
    lsA[row * 68 + c4 + 2] = v.z; lsA[row * 68 + c4 + 3] = v.w;
  }
#ifdef USE_TDM
  if (wave == 0) __builtin_amdgcn_s_wait_tensorcnt(0);
#endif
  __syncthreads();

  const int r = lane & 15, kb = (lane >> 4) * 2;
  v8f acc = {0,0,0,0,0,0,0,0};
#pragma unroll
  for (int k = 0; k < 64; k += 4) {
    v2f a; a.x = lsA[r * 68 + k + kb]; a.y = lsA[r * 68 + k + kb + 1];
    int col = wave * 16 + r;
    v2f bv;
    bv.x = lsB[col * 68 + k + kb];
    bv.y = lsB[col * 68 + k + kb + 1];
    acc = wmma4(a, bv, acc);
  }
  const int hi = lane >> 4, col = lane & 15;
  if (hi == 0) { // rows 8..15 were zero-padded
#pragma unroll
    for (int rr = 0; rr < 8; ++rr) {
      long idx = (((long)(b * 8 + rr) * 512 + l) * 512) + m0 + wave * 16 + col;
      scores[idx] += acc[rr];
    }
  }
}

// In-place masked softmax on attn rows (scores live in d_out attn region).
__global__ __launch_bounds__(256) void k_softmax(
    float* __restrict__ attn, const int* __restrict__ adj) {
  __shared__ float red[256];
  const long row = blockIdx.x;            // (b*8+h)*512 + l
  const int l = row & 511, b = (int)(row >> 12);
  float* p = attn + row * 512;
  const int* am = adj + ((long)(b * 512 + l)) * 512;
  const int t = threadIdx.x;
  float v0 = (am[t] == 0) ? -10000.0f : p[t];
  float v1 = (am[t + 256] == 0) ? -10000.0f : p[t + 256];
  red[t] = fmaxf(v0, v1);
  __syncthreads();
  for (int s = 128; s > 0; s >>= 1) {
    if (t < s) red[t] = fmaxf(red[t], red[t + s]);
    __syncthreads();
  }
  float mx = red[0];
  __syncthreads();
  float e0 = __expf(v0 - mx), e1 = __expf(v1 - mx);
  red[t] = e0 + e1;
  __syncthreads();
  for (int s = 128; s > 0; s >>= 1) {
    if (t < s) red[t] += red[t + s];
    __syncthreads();
  }
  float inv = 1.0f / red[0];
  p[t] = e0 * inv;
  p[t + 256] = e1 * inv;
}

// Relative output: per (b,l): O_rel[h,d] = sum_m attn[b,h,l,m]*vec_adj_v[b,l,m,d]
// B tile (32 x 64 f32 per K stage) DMA'd by TDM, pad => LDS stride 68.
__global__ __launch_bounds__(128) void k_rel_out(
    const float* __restrict__ attn, const float* __restrict__ vav,
    float* __restrict__ qc) {
  __shared__ float lsA[16 * 36];
  __shared__ float lsB[32 * 68];
  const int bl = blockIdx.x, b = bl >> 9, l = bl & 511;
  const int t = threadIdx.x, lane = t & 31, wave = t >> 5;  // 4 waves
  const int r = lane & 15, kb = (lane >> 4) * 2;
  v8f acc = {0,0,0,0,0,0,0,0};
  for (int k0 = 0; k0 < 512; k0 += 32) {
    __syncthreads();
#ifdef USE_TDM
    if (wave == 0) {
      tdm_load_2d((unsigned)(size_t)(void*)lsB,
                  vav + ((long)bl * 512 + k0) * 64,
                  /*tensor_d0=*/64, /*tensor_d1=*/512,
                  /*tile_d0=*/64, /*tile_d1=*/32, /*d0_stride=*/64);
    }
#else
    { // B: vec_adj_v[b,l, k0+kr, :64]
      int kr = t >> 2;
      const float* src = vav + ((long)bl * 512 + k0 + kr) * 64;
#pragma unroll
      for (int j = 0; j < 4; ++j) {
        int c = (t & 3) * 4 + j * 16;
        float4 v = *(const float4*)(src + c);
        *(float4*)(lsB + kr * 68 + c) = v;
      }
    }
#endif
    { // A: attn rows h (pad to 16), 32 K-cols
      int row = t >> 3, c4 = (t & 7) * 4;
      float4 v = {0.f, 0.f, 0.f, 0.f};
      if (row < 8)
        v = *(const float4*)(attn + ((long)(b * 8 + row) * 512 + l) * 512 + k0 + c4);
      lsA[row * 36 + c4 + 0] = v.x; lsA[row * 36 + c4 + 1] = v.y;
      lsA[row * 36 + c4 + 2] = v.z; lsA[row * 36 + c4 + 3] = v.w;
    }
#ifdef USE_TDM
    if (wave == 0) __builtin_amdgcn_s_wait_tensorcnt(0);
#endif
    __syncthreads();
#pragma unroll
    for (int k = 0; k < 32; k += 4) {
      v2f a; a.x = lsA[r * 36 + k + kb]; a.y = lsA[r * 36 + k + kb + 1];
      v2f bv; int col = wave * 16 + r;
      bv.x = lsB[(k + kb) * 68 + col];
      bv.y = lsB[(k + kb + 1) * 68 + col];
      acc = wmma4(a, bv, acc);
    }
  }
  const int hi = lane >> 4, col = lane & 15;
  if (hi == 0) {
#pragma unroll
    for (int rr = 0; rr < 8; ++rr)
      qc[(long)bl * 512 + rr * 64 + wave * 16 + col] += acc[rr];
  }
}

// node_weight = sigmoid(qc @ w_nw); denom = max(row-sum(adj), 1)
__global__ __launch_bounds__(256) void k_node(
    const float* __restrict__ qc, const float* __restrict__ w_nw,
    const int* __restrict__ adj, float* __restrict__ nw,
    float* __restrict__ denom) {
  __shared__ float r1[256];
  __shared__ float r2[256];
  const int bl = blockIdx.x, t = threadIdx.x;
  const float* qr = qc + (long)bl * 512;
  const int* ar = adj + (long)bl * 512;
  r1[t] = qr[t] * w_nw[t] + qr[t + 256] * w_nw[t + 256];
  r2[t] = (float)ar[t] + (float)ar[t + 256];
  __syncthreads();
  for (int s = 128; s > 0; s >>= 1) {
    if (t < s) { r1[t] += r1[t + s]; r2[t] += r2[t + s]; }
    __syncthreads();
  }
  if (t == 0) {
    nw[bl] = 1.0f / (1.0f + __expf(-r1[0]));
    denom[bl] = fmaxf(r2[0], 1.0f);
  }
}

extern "C" void kernel_launch(void* const* d_in, const int* in_sizes, int n_in,
                              void* d_out, int out_size, void* d_ws, size_t ws_size,
                              hipStream_t stream) {
  const float* q    = (const float*)d_in[0];
  const float* k    = (const float*)d_in[1];
  const float* v    = (const float*)d_in[2];
  const float* vak  = (const float*)d_in[3];
  const float* vav  = (const float*)d_in[4];
  const int*   adjk = (const int*)d_in[5];
  // d_in[6] (adj_v) unused by the reference
  const float* w_qs = (const float*)d_in[7];
  const float* w_ks = (const float*)d_in[8];
  const float* w_vs = (const float*)d_in[9];
  const float* w_fc = (const float*)d_in[10];
  const float* w_nw = (const float*)d_in[11];
  const float* w_self = (const float*)d_in[12];
  const float* w_dd = (const float*)d_in[13];

  const long M1 = 1048576;  // 2048*512
  float* ws = (float*)d_ws;
  float* qs_p = ws;            // [2048,512]  q proj, pre-scaled 1/sqrt(Dk)
  float* kh_p = qs_p + M1;     // [2048,512]
  float* vh_p = kh_p + M1;     // [2048,512]
  float* khT  = vh_p + M1;     // [b][h][64][512]
  float* qc   = khT + M1;      // [2048,512]
  float* self_info = qc + M1;  // [2048,512]
  float* dd_info = self_info + M1;
  float* xbuf = dd_info + M1;
  float* nw = xbuf + M1;       // [2048]
  float* denom = nw + 2048;    // [2048]

  float* qo = (float*)d_out;       // [2048,512]
  float* attn = qo + M1;           // [4,8,512,512] — also used as scores scratch

  dim3 blk(256);
  const long S_bh = 512L * 512L;

  // 1) projections (q scaled by 1/temperature = 1/8)
  k_gemm_batched<<<dim3(8, 32, 1), blk, 0, stream>>>(q, 0, 0, 512, w_qs, 0, 0, 512,
                                                     qs_p, 0, 0, 512, 512, 0.125f);
  k_gemm_batched<<<dim3(8, 32, 1), blk, 0, stream>>>(k, 0, 0, 512, w_ks, 0, 0, 512,
                                                     kh_p, 0, 0, 512, 512, 1.0f);
  k_gemm_batched<<<dim3(8, 32, 1), blk, 0, stream>>>(v, 0, 0, 512, w_vs, 0, 0, 512,
                                                     vh_p, 0, 0, 512, 512, 1.0f);
  // 2) kh transpose for row-major content-score B
  k_transpose_kh<<<4096, 256, 0, stream>>>(kh_p, khT);
  // 3) content scores -> attn region (z = b*8+h)
  k_gemm_batched<<<dim3(8, 8, 32), blk, 0, stream>>>(
      qs_p, S_bh, 64, 512, khT, 8L * 64 * 512, 64L * 512, 512,
      attn, 8L * S_bh, S_bh, 512, 64, 1.0f);
  // 4) relative-position scores (streams vec_adj_k, 256MB, via TDM)
  k_rel_scores<<<dim3(2048, 4), blk, 0, stream>>>(qs_p, vak, attn);
  // 5) masked softmax in place
  k_softmax<<<16384, 256, 0, stream>>>(attn, adjk);
  // 6) content output -> qc [b*512+l][h*64+d]
  k_gemm_batched<<<dim3(1, 8, 32), blk, 0, stream>>>(
      attn, 8L * S_bh, S_bh, 512, vh_p, S_bh, 64, 512,
      qc, S_bh, 64, 512, 512, 1.0f);
  // 7) relative output (streams vec_adj_v, 256MB, via TDM)
  k_rel_out<<<2048, 128, 0, stream>>>(attn, vav, qc);
  // 8) node weights + denominators
  k_node<<<2048, 256, 0, stream>>>(qc, w_nw, adjk, nw, denom);
  // 9) self/dd projections
  k_gemm_batched<<<dim3(8, 32, 1), blk, 0, stream>>>(qc, 0, 0, 512, w_self, 0, 0, 512,
                                                     self_info, 0, 0, 512, 512, 1.0f);
  k_gemm_batched<<<dim3(8, 32, 1), blk, 0, stream>>>(qc, 0, 0, 512, w_dd, 0, 0, 512,
                                                     dd_info, 0, 0, 512, 512, 1.0f);
  // 10) masked neighbor aggregation
  k_agg<<<dim3(8, 8, 4), blk, 0, stream>>>(adjk, nw, dd_info, self_info, denom, xbuf);
  // 11) FC + GELU + residual -> qo
  k_fc<<<dim3(8, 32, 1), blk, 0, stream>>>(xbuf, w_fc, q, qo);
}